// GraphLearner_50371376447612
// MI455X (gfx1250) — compile-verified
//
#include <hip/hip_runtime.h>

// GAT-style attention map:
//   u1 = w @ a[:H], u2 = w @ a[H:]            (fold: 512x512 GEMV x2, ~1 MFLOP)
//   f1 = context @ u1, f2 = context @ u2      (WMMA fp32, 67 MFLOP)
//   out[b,i,j] = softplus(softmax_j(mask(leaky_relu(f1[b,i]+f2[b,j]))))
// Bandwidth bound: ~335 MB HBM traffic -> ~14 us at 23.3 TB/s.

#define ALPHA 0.2f
#define NEG_BIG -9.0e15f

static constexpr int Bn  = 32;
static constexpr int Nn  = 1024;
static constexpr int Fin = 512;
static constexpr int Hh  = 512;

typedef float v2f __attribute__((ext_vector_type(2)));
typedef float v8f __attribute__((ext_vector_type(8)));

// ---------------------------------------------------------------------------
// Kernel 1: fold the attention vector through w.
//   u[f]       = sum_h w[f,h] * a[h]
//   u[Fin + f] = sum_h w[f,h] * a[Hh + h]
// w rows are contiguous per thread -> sequential per-thread streaming, 1 MB.
// ---------------------------------------------------------------------------
__global__ void gat_fold_w(const float* __restrict__ w,
                           const float* __restrict__ a,
                           float* __restrict__ u) {
  const int f = blockIdx.x * blockDim.x + threadIdx.x;
  if (f >= Fin) return;
  const float* wr = w + (size_t)f * Hh;
  float s1 = 0.f, s2 = 0.f;
  for (int h = 0; h < Hh; ++h) {
    const float wv = wr[h];
    s1 = fmaf(wv, a[h], s1);
    s2 = fmaf(wv, a[Hh + h], s2);
  }
  u[f] = s1;
  u[Fin + f] = s2;
}

// ---------------------------------------------------------------------------
// Kernel 2: f1/f2 = context @ [u1|u2] via V_WMMA_F32_16X16X4_F32 (full fp32).
// One wave32 owns a 16-row tile of context (rows are flattened b*N+n).
// A layout (16x4 f32): lane L -> (M = L&15, K = 2*(L>>4) + {0,1})  = float2.
// B layout (4x16 f32): lane L -> (N = L&15, K = 2*(L>>4) + {0,1});
//   column N=0 carries u1, N=1 carries u2, all other columns are zero.
// C/D (16x16 f32): VGPR v, lanes 0-15 -> (M=v, N=lane); lanes 16-31 -> (M=v+8).
// 128 WMMAs accumulate K=512; EXEC stays all-ones (branch-free inner loop:
// every lane streams one u-vector unconditionally and zeros unused columns
// with a 0/1 multiplier -> no predicated loads, no saveexec in the hot loop).
// ---------------------------------------------------------------------------
__global__ __launch_bounds__(256) void gat_f12_wmma(
    const float* __restrict__ ctx, const float* __restrict__ u,
    float* __restrict__ f1, float* __restrict__ f2) {
  const int lane  = threadIdx.x & 31;
  const int wave  = threadIdx.x >> 5;
  const int tile  = blockIdx.x * (blockDim.x >> 5) + wave; // 16-row tile id
  const long row0 = (long)tile * 16;

  const int m     = lane & 15;        // M (for A) / N (for B) within half-wave
  const int khalf = (lane >> 4) << 1; // 0 or 2: K sub-offset carried by lane

  const float* __restrict__ arow = ctx + (row0 + m) * (long)Fin + khalf;
  // Branch-free B operand: lane's source vector + 0/1 column mask.
  const float* __restrict__ bptr = ((m == 1) ? (u + Fin) : u) + khalf;
  const float bsel = (m <= 1) ? 1.f : 0.f;

  v8f c = {0.f, 0.f, 0.f, 0.f, 0.f, 0.f, 0.f, 0.f};

#pragma unroll 4
  for (int k = 0; k < Fin; k += 4) {
    // A: two consecutive K values for this lane's row (8B aligned).
    const v2f av = *reinterpret_cast<const v2f*>(arow + k);
    // B: unconditional aligned load, masked to columns N in {0,1}.
    const v2f braw = *reinterpret_cast<const v2f*>(bptr + k);
    v2f bv;
    bv.x = braw.x * bsel;
    bv.y = braw.y * bsel;
    c = __builtin_amdgcn_wmma_f32_16x16x4_f32(
        /*neg_a=*/false, av, /*neg_b=*/false, bv,
        /*c_mod=*/(short)0, c, /*reuse_a=*/false, /*reuse_b=*/false);
  }

  // Write-out: column 0 of D is f1, column 1 is f2.
  if (m <= 1) {
    float* __restrict__ dst = (m == 0) ? f1 : f2;
    const long base = row0 + (long)((lane >> 4) << 3); // lanes<16: M=0..7, else M=8..15
#pragma unroll
    for (int v = 0; v < 8; ++v) dst[base + v] = c[v];
  }
}

// ---------------------------------------------------------------------------
// Kernel 3: one 256-thread block per output row (b,i).
// Each thread keeps 4 of the 1024 row elements in registers:
//   e = mask(leaky_relu(f1[b,i] + f2[b,j])); softmax over j; softplus.
// 128-bit adj loads, 128-bit out stores; wave32 shfl_xor + 8-slot LDS reduce.
// All-masked rows degrade to uniform 1/N exactly like jax softmax.
// ---------------------------------------------------------------------------
__global__ __launch_bounds__(256) void gat_softmax_row(
    const int* __restrict__ adj, const float* __restrict__ f1,
    const float* __restrict__ f2, float* __restrict__ out) {
  __shared__ float sm[8];

  const int row  = blockIdx.x;      // b*N + i
  const int b    = row >> 10;       // N == 1024
  const int tid  = threadIdx.x;
  const int lane = tid & 31;
  const int wv   = tid >> 5;

  const float s = f1[row];

  const int4   aj = reinterpret_cast<const int4*>(adj + (size_t)row * Nn)[tid];
  const float4 g  = reinterpret_cast<const float4*>(f2 + (size_t)b * Nn)[tid];

  auto lrelu = [](float x) { return x > 0.f ? x : ALPHA * x; };
  float e[4];
  e[0] = (aj.x > 0) ? lrelu(s + g.x) : NEG_BIG;
  e[1] = (aj.y > 0) ? lrelu(s + g.y) : NEG_BIG;
  e[2] = (aj.z > 0) ? lrelu(s + g.z) : NEG_BIG;
  e[3] = (aj.w > 0) ? lrelu(s + g.w) : NEG_BIG;

  // ---- row max ----
  float mx = fmaxf(fmaxf(e[0], e[1]), fmaxf(e[2], e[3]));
#pragma unroll
  for (int off = 16; off > 0; off >>= 1)
    mx = fmaxf(mx, __shfl_xor(mx, off, 32));
  if (lane == 0) sm[wv] = mx;
  __syncthreads();
  {
    float t = sm[0];
#pragma unroll
    for (int i = 1; i < 8; ++i) t = fmaxf(t, sm[i]);
    mx = t;
  }
  __syncthreads(); // sm reused below

  // ---- exp + row sum ----
  float p[4];
  float ls = 0.f;
#pragma unroll
  for (int i = 0; i < 4; ++i) {
    p[i] = __expf(e[i] - mx); // masked terms underflow to exactly 0
    ls += p[i];
  }
#pragma unroll
  for (int off = 16; off > 0; off >>= 1)
    ls += __shfl_xor(ls, off, 32);
  if (lane == 0) sm[wv] = ls;
  __syncthreads();
  float sum = sm[0];
#pragma unroll
  for (int i = 1; i < 8; ++i) sum += sm[i];
  const float inv = 1.f / sum;

  // ---- softplus(attention) and 128-bit store ----
  float4 o;
  o.x = __logf(1.f + __expf(p[0] * inv));
  o.y = __logf(1.f + __expf(p[1] * inv));
  o.z = __logf(1.f + __expf(p[2] * inv));
  o.w = __logf(1.f + __expf(p[3] * inv));
  reinterpret_cast<float4*>(out + (size_t)row * Nn)[tid] = o;
}

// ---------------------------------------------------------------------------
extern "C" void kernel_launch(void* const* d_in, const int* in_sizes, int n_in,
                              void* d_out, int out_size, void* d_ws, size_t ws_size,
                              hipStream_t stream) {
  const float* ctx = (const float*)d_in[0]; // (32,1024,512)
  const int*   adj = (const int*)d_in[1];   // (32,1024,1024)
  const float* w   = (const float*)d_in[2]; // (512,512)
  const float* a   = (const float*)d_in[3]; // (1024,1)
  float* out = (float*)d_out;               // (32,1024,1024)

  float* u  = (float*)d_ws;        // 1024 floats: [u1 | u2]
  float* f1 = u + 2 * Fin;         // 32768 floats
  float* f2 = f1 + Bn * Nn;        // 32768 floats

  gat_fold_w<<<(Fin + 255) / 256, 256, 0, stream>>>(w, a, u);

  const int tiles = (Bn * Nn) / 16;            // 2048 wave-tiles
  gat_f12_wmma<<<tiles / 8, 256, 0, stream>>>(ctx, u, f1, f2);

  gat_softmax_row<<<Bn * Nn, 256, 0, stream>>>(adj, f1, f2, out);
}